// MOELayer_53051436040496
// MI455X (gfx1250) — compile-verified
//
#include <hip/hip_runtime.h>
#include <math.h>

// Problem constants (match reference)
#define B_    2
#define S_    512
#define H_    1024
#define E_    4
#define V_    32000
#define NTOK  (B_ * S_)          // 1024 tokens
#define THRESH 0.5f

typedef float v2f __attribute__((ext_vector_type(2)));
typedef float v4f __attribute__((ext_vector_type(4)));
typedef float v8f __attribute__((ext_vector_type(8)));

// ---------------------------------------------------------------------------
// Kernel 1: router.  One wave (32 threads) per 16-token tile.
// route/noise logits via V_WMMA_F32_16X16X4_F32 (exact fp32 GEMM).
// Weights are staged TRANSPOSED into LDS so each lane's B operand per k-step
// is a contiguous, 8B-aligned pair -> single ds_load_b64, no divergence
// (EXEC stays all-1s, as WMMA requires).  N>=4 padding is a mask multiply.
//
// WMMA f32 16x16x4 VGPR layout (ISA 7.12.2):
//   A (16x4, MxK):  lane L holds row M=L&15; half=(L>>4): VGPR j -> K = 2*half + j
//   B (4x16, KxN):  lane L holds col N=L&15; half=(L>>4): VGPR j -> K = 2*half + j
//   C/D (16x16):    lane L holds col N=L&15; VGPR j -> M = j + 8*(L>>4)
// ---------------------------------------------------------------------------
__global__ __launch_bounds__(32)
void router_kernel(const float* __restrict__ emb,     // [NTOK, H]
                   const float* __restrict__ noise,   // [NTOK, E]
                   const float* __restrict__ Wr,      // [H, E]
                   const float* __restrict__ br,      // [E]
                   const float* __restrict__ Wn,      // [H, E]
                   const float* __restrict__ bn,      // [E]
                   float* __restrict__ out_prob1,     // [NTOK, E]
                   float* __restrict__ out_prob2,     // [NTOK, E]
                   float* __restrict__ ws_sp,         // [NTOK, E] sorted probs
                   int*   __restrict__ ws_si,         // [NTOK, E] sorted idx
                   int*   __restrict__ ws_k)          // [1] global max_k
{
    const int lane  = threadIdx.x;      // 0..31
    const int tile  = blockIdx.x;       // 0..NTOK/16-1
    const int row16 = lane & 15;
    const int half  = lane >> 4;        // 0 or 1

    // Stage transposed weights: WrT[e][h] = Wr[h][e].  16KB each, 32KB total.
    __shared__ float WrT[E_][H_];
    __shared__ float WnT[E_][H_];
    for (int idx = lane; idx < H_ * E_; idx += 32) {   // uniform trip count
        const int h = idx >> 2;
        const int e = idx & 3;
        WrT[e][h] = Wr[idx];
        WnT[e][h] = Wn[idx];
    }
    __syncthreads();

    const int   rclamp = (row16 < E_) ? row16 : 0;     // lanes >=4 read row 0
    const float bmask  = (row16 < E_) ? 1.0f : 0.0f;   // ...and zero it out

    v8f cR = {};
    v8f cN = {};

    // A row pointer: row = tile*16 + row16, K base 2*half (8B aligned)
    const float* arow = emb + (size_t)(tile * 16 + row16) * H_ + 2 * half;
    const float* brow = &WrT[rclamp][2 * half];
    const float* nrow = &WnT[rclamp][2 * half];

    for (int k0 = 0; k0 < H_; k0 += 4) {
        v2f a  = *(const v2f*)(arow + k0);   // global_load_b64
        v2f bR = *(const v2f*)(brow + k0);   // ds_load_b64
        v2f bN = *(const v2f*)(nrow + k0);   // ds_load_b64
        bR *= bmask;
        bN *= bmask;

        // 8 args: (neg_a, A, neg_b, B, c_mod, C, reuse_a, reuse_b)
        cR = __builtin_amdgcn_wmma_f32_16x16x4_f32(false, a, false, bR,
                                                   (short)0, cR, false, false);
        cN = __builtin_amdgcn_wmma_f32_16x16x4_f32(false, a, false, bN,
                                                   (short)0, cN, false, false);
    }

    // Deposit the 16x4 valid region into LDS for per-token processing.
    __shared__ float sR[16][E_];
    __shared__ float sN[16][E_];
    if (row16 < E_) {
        #pragma unroll
        for (int j = 0; j < 8; ++j) {
            const int m = j + 8 * half;        // token row within tile
            sR[m][row16] = cR[j];
            sN[m][row16] = cN[j];
        }
    }
    __syncthreads();

    if (lane < 16) {
        const int t = tile * 16 + lane;        // global token id

        // Noisy gating logits
        float l[E_];
        float mx = -INFINITY;
        #pragma unroll
        for (int e = 0; e < E_; ++e) {
            const float rl = sR[lane][e] + br[e];
            const float nl = sN[lane][e] + bn[e];
            const float sp = (nl > 20.0f) ? nl : log1pf(expf(nl)); // softplus
            l[e] = rl + noise[(size_t)t * E_ + e] * sp;
            mx = fmaxf(mx, l[e]);
        }

        // Softmax over E=4
        float p[E_];
        float sum = 0.0f;
        #pragma unroll
        for (int e = 0; e < E_; ++e) { p[e] = expf(l[e] - mx); sum += p[e]; }
        const float inv = 1.0f / sum;
        #pragma unroll
        for (int e = 0; e < E_; ++e) {
            p[e] *= inv;
            out_prob1[(size_t)t * E_ + e] = p[e];  // route_prob
            out_prob2[(size_t)t * E_ + e] = p[e];  // route_prob_revised
        }

        // Sort 4 descending with index tracking (comparator network)
        float v[E_] = {p[0], p[1], p[2], p[3]};
        int   id[E_] = {0, 1, 2, 3};
        #define CSWAP(a, b)                                              \
            if (v[a] < v[b]) {                                           \
                float tv_ = v[a]; v[a] = v[b]; v[b] = tv_;               \
                int   ti_ = id[a]; id[a] = id[b]; id[b] = ti_;           \
            }
        CSWAP(0, 1) CSWAP(2, 3) CSWAP(0, 2) CSWAP(1, 3) CSWAP(1, 2)
        #undef CSWAP

        // cumsum threshold; mask[0] forced true
        int k = 0;
        float cum = 0.0f;
        #pragma unroll
        for (int i = 0; i < E_; ++i) {
            cum += v[i];
            if (i == 0 || cum < THRESH) ++k;
        }

        #pragma unroll
        for (int i = 0; i < E_; ++i) {
            ws_sp[(size_t)t * E_ + i] = v[i];
            ws_si[(size_t)t * E_ + i] = id[i];
        }
        atomicMax(ws_k, k);
    }
}

// ---------------------------------------------------------------------------
// Kernel 2: combine.  out[t, v] = sum_e w_e(t) * logits[e, t, v].
// Pure HBM streaming: ~655 MB moved -> ~28 us floor at 23.3 TB/s.
// Non-temporal b128 loads/stores (logits working set 524 MB >> 192 MB L2).
// grid = (ceil(V/4 / 256), NTOK), block = 256.
// ---------------------------------------------------------------------------
__global__ __launch_bounds__(256)
void combine_kernel(const float* __restrict__ logits,  // [E, NTOK, V]
                    const float* __restrict__ ws_sp,
                    const int*   __restrict__ ws_si,
                    const int*   __restrict__ ws_k,
                    float* __restrict__ out)            // [NTOK, V]
{
    const int t = blockIdx.y;
    const int k = *ws_k;

    // Reconstruct per-expert weights in original order (cheap, L2-resident).
    float sp[E_];
    int   si[E_];
    #pragma unroll
    for (int i = 0; i < E_; ++i) {
        sp[i] = ws_sp[(size_t)t * E_ + i];
        si[i] = ws_si[(size_t)t * E_ + i];
    }
    float sum = 0.0f;
    #pragma unroll
    for (int i = 0; i < E_; ++i) if (i < k) sum += sp[i];
    const float inv = 1.0f / (sum + 1e-6f);
    float we[E_] = {0.0f, 0.0f, 0.0f, 0.0f};
    #pragma unroll
    for (int i = 0; i < E_; ++i) if (i < k) we[si[i]] = sp[i] * inv;

    const int n4 = V_ / 4;                       // 8000 float4 per token
    const int i4 = blockIdx.x * blockDim.x + threadIdx.x;
    if (i4 < n4) {
        const size_t tv = (size_t)t * V_;
        const v4f* l0 = (const v4f*)(logits + 0 * (size_t)NTOK * V_ + tv);
        const v4f* l1 = (const v4f*)(logits + 1 * (size_t)NTOK * V_ + tv);
        const v4f* l2 = (const v4f*)(logits + 2 * (size_t)NTOK * V_ + tv);
        const v4f* l3 = (const v4f*)(logits + 3 * (size_t)NTOK * V_ + tv);

        const v4f a0 = __builtin_nontemporal_load(&l0[i4]);
        const v4f a1 = __builtin_nontemporal_load(&l1[i4]);
        const v4f a2 = __builtin_nontemporal_load(&l2[i4]);
        const v4f a3 = __builtin_nontemporal_load(&l3[i4]);

        const v4f r = we[0] * a0 + we[1] * a1 + we[2] * a2 + we[3] * a3;
        __builtin_nontemporal_store(r, (v4f*)(out + tv) + i4);
    }
}

// ---------------------------------------------------------------------------
extern "C" void kernel_launch(void* const* d_in, const int* in_sizes, int n_in,
                              void* d_out, int out_size, void* d_ws, size_t ws_size,
                              hipStream_t stream)
{
    const float* emb    = (const float*)d_in[0];  // [B,S,H]
    const float* logits = (const float*)d_in[1];  // [E,B,S,V]
    const float* noise  = (const float*)d_in[2];  // [B,S,E]
    const float* Wr     = (const float*)d_in[3];  // [H,E]
    const float* br     = (const float*)d_in[4];  // [E]
    const float* Wn     = (const float*)d_in[5];  // [H,E]
    const float* bn     = (const float*)d_in[6];  // [E]

    float* out       = (float*)d_out;
    float* out_prob1 = out + (size_t)NTOK * V_;           // route_prob
    float* out_prob2 = out_prob1 + (size_t)NTOK * E_;     // route_prob_revised

    // Workspace layout: [0..4) max_k int, [64..) sorted_prob, then sorted_idx
    char*  ws    = (char*)d_ws;
    int*   ws_k  = (int*)ws;
    float* ws_sp = (float*)(ws + 64);
    int*   ws_si = (int*)(ws + 64 + (size_t)NTOK * E_ * sizeof(float));

    hipMemsetAsync(ws_k, 0, sizeof(int), stream);

    router_kernel<<<dim3(NTOK / 16), dim3(32), 0, stream>>>(
        emb, noise, Wr, br, Wn, bn,
        out_prob1, out_prob2, ws_sp, ws_si, ws_k);

    const int n4 = V_ / 4;
    combine_kernel<<<dim3((n4 + 255) / 256, NTOK), dim3(256), 0, stream>>>(
        logits, ws_sp, ws_si, ws_k, out);
}